// SingleQubitOperationLayer_60507499266598
// MI455X (gfx1250) — compile-verified
//
#include <hip/hip_runtime.h>
#include <math.h>

#define NQ     12
#define DIM    4096          // 2^12
#define BATCH  2048
#define WPB    8             // waves per block (1 row per wave)
#define PLANE  4368          // 4096 + 256 + 16 padded floats per LDS plane

typedef __attribute__((ext_vector_type(2))) float v2f;
typedef __attribute__((ext_vector_type(8))) float v8f;

// LDS anti-bank-conflict padding (injective, max 4365 < PLANE)
__device__ __forceinline__ int pidx(int a) { return a + (a >> 4) + (a >> 8); }

// D(16x16,f32) = A(16x4,f32) x B(4x16,f32) + C  -- CDNA5 WMMA
__device__ __forceinline__ v8f wmma4(v2f a, v2f b, v8f c) {
  return __builtin_amdgcn_wmma_f32_16x16x4_f32(false, a, false, b, (short)0, c,
                                               false, false);
}

// Constant 4-qubit beamsplitter M16[i][j] = 0.25 * i^popcount(i^j), held as
// A-fragments (M = lane&15, K = 4q + v + 2*(lane>>4)). M16 is symmetric.
struct MFrag { v2f r[4]; v2f i[4]; v2f ni[4]; };

__device__ __forceinline__ MFrag make_m16(int lane) {
  MFrag f;
  const int lm = lane & 15, lh = lane >> 4;
#pragma unroll
  for (int q = 0; q < 4; ++q) {
#pragma unroll
    for (int v = 0; v < 2; ++v) {
      const int k = 4 * q + v + 2 * lh;
      const int h = __popc(k ^ lm) & 3;
      const float re = (h == 0) ? 0.25f : ((h == 2) ? -0.25f : 0.0f);
      const float im = (h == 1) ? 0.25f : ((h == 3) ? -0.25f : 0.0f);
      if (v == 0) { f.r[q].x = re; f.i[q].x = im; f.ni[q].x = -im; }
      else        { f.r[q].y = re; f.i[q].y = im; f.ni[q].y = -im; }
    }
  }
  return f;
}

// Complex 16x16 (K=16) product: D = M16 * Bx  (M16 const complex, Bx data)
__device__ __forceinline__ void cwmma(const MFrag& A, const v2f* bxr,
                                      const v2f* bxi, v8f& dr, v8f& di) {
  const v8f z = {0.f, 0.f, 0.f, 0.f, 0.f, 0.f, 0.f, 0.f};
  dr = z; di = z;
#pragma unroll
  for (int q = 0; q < 4; ++q) dr = wmma4(A.r[q],  bxr[q], dr);
#pragma unroll
  for (int q = 0; q < 4; ++q) dr = wmma4(A.ni[q], bxi[q], dr);  // -Im(M16)*Im(x)
#pragma unroll
  for (int q = 0; q < 4; ++q) di = wmma4(A.i[q],  bxr[q], di);
#pragma unroll
  for (int q = 0; q < 4; ++q) di = wmma4(A.r[q],  bxi[q], di);
}

// Generic LDS->LDS axis pass. B addr = t*TB + k*KB + lm*NB ;
// D addr = t*TB + M*MD + lm*ND. PHASE: fold elementwise phase (column == linear
// address, valid for the P1-style pass only).
template <int TB, int KB, int NB, int MD, int ND, bool PHASE>
__device__ __forceinline__ void lds_pass(const MFrag& A, float* xr, float* xi,
                                         const float* __restrict__ phr,
                                         const float* __restrict__ phi,
                                         int lane) {
  const int lm = lane & 15, lh = lane >> 4;
  for (int t = 0; t < 16; ++t) {
    v2f bxr[4], bxi[4];
#pragma unroll
    for (int q = 0; q < 4; ++q) {
      const int a0 = t * TB + (4 * q + 2 * lh) * KB + lm * NB;
      float r0 = xr[pidx(a0)], r1 = xr[pidx(a0 + KB)];
      float i0 = xi[pidx(a0)], i1 = xi[pidx(a0 + KB)];
      if (PHASE) {
        const float cr0 = phr[a0], ci0 = phi[a0];
        const float cr1 = phr[a0 + KB], ci1 = phi[a0 + KB];
        const float t0r = r0 * cr0 - i0 * ci0, t0i = r0 * ci0 + i0 * cr0;
        const float t1r = r1 * cr1 - i1 * ci1, t1i = r1 * ci1 + i1 * cr1;
        r0 = t0r; i0 = t0i; r1 = t1r; i1 = t1i;
      }
      bxr[q].x = r0; bxr[q].y = r1;
      bxi[q].x = i0; bxi[q].y = i1;
    }
    v8f dr, di;
    cwmma(A, bxr, bxi, dr, di);
#pragma unroll
    for (int r = 0; r < 8; ++r) {
      const int d = t * TB + (r + 8 * lh) * MD + lm * ND;
      xr[pidx(d)] = dr[r];
      xi[pidx(d)] = di[r];
    }
  }
}

// First pass: B fragments straight from global x (fold d_in), D -> LDS.
__device__ __forceinline__ void pass1_global(const MFrag& A,
    const float* __restrict__ xrg, const float* __restrict__ xig,
    const float* __restrict__ cr, const float* __restrict__ ci,
    float* xr, float* xi, int lane) {
  const int lm = lane & 15, lh = lane >> 4;
  for (int t = 0; t < 16; ++t) {
    const int base = t * 256 + lm * 16;
    v2f bxr[4], bxi[4];
#pragma unroll
    for (int q = 0; q < 4; ++q) {
      const float4 r4 = *(const float4*)(xrg + base + 4 * q);
      const float4 i4 = *(const float4*)(xig + base + 4 * q);
      const float4 c4 = *(const float4*)(cr + base + 4 * q);
      const float4 s4 = *(const float4*)(ci + base + 4 * q);
      const float r0 = lh ? r4.z : r4.x, r1 = lh ? r4.w : r4.y;
      const float i0 = lh ? i4.z : i4.x, i1 = lh ? i4.w : i4.y;
      const float c0 = lh ? c4.z : c4.x, c1 = lh ? c4.w : c4.y;
      const float s0 = lh ? s4.z : s4.x, s1 = lh ? s4.w : s4.y;
      bxr[q].x = r0 * c0 - i0 * s0; bxi[q].x = r0 * s0 + i0 * c0;
      bxr[q].y = r1 * c1 - i1 * s1; bxi[q].y = r1 * s1 + i1 * c1;
    }
    v8f dr, di;
    cwmma(A, bxr, bxi, dr, di);
#pragma unroll
    for (int r = 0; r < 8; ++r) {
      const int d = t * 256 + lm * 16 + (r + 8 * lh);
      xr[pidx(d)] = dr[r];
      xi[pidx(d)] = di[r];
    }
  }
}

// Last pass: B from LDS, fold d_ph into D, store interleaved complex to global.
__device__ __forceinline__ void pass3_global(const MFrag& A, float* xr, float* xi,
    const float* __restrict__ cphr, const float* __restrict__ cphi,
    float* __restrict__ outg, int lane) {
  const int lm = lane & 15, lh = lane >> 4;
  for (int t = 0; t < 16; ++t) {
    v2f bxr[4], bxi[4];
#pragma unroll
    for (int q = 0; q < 4; ++q) {
      const int a0 = (4 * q + 2 * lh) * 256 + t * 16 + lm;
      bxr[q].x = xr[pidx(a0)]; bxr[q].y = xr[pidx(a0 + 256)];
      bxi[q].x = xi[pidx(a0)]; bxi[q].y = xi[pidx(a0 + 256)];
    }
    v8f dr, di;
    cwmma(A, bxr, bxi, dr, di);
#pragma unroll
    for (int r = 0; r < 8; ++r) {
      const int c = (r + 8 * lh) * 256 + t * 16 + lm;   // final column index
      const float pr = cphr[c], pi = cphi[c];
      float2 o;
      o.x = dr[r] * pr - di[r] * pi;
      o.y = dr[r] * pi + di[r] * pr;
      *(float2*)(outg + 2 * c) = o;
    }
  }
}

__global__ void __launch_bounds__(32 * WPB) qphot_kernel(
    const float* __restrict__ x_real, const float* __restrict__ x_imag,
    const float* __restrict__ ph, float* __restrict__ out) {
  extern __shared__ float lds[];
  const int lane = threadIdx.x & 31;
  const int wave = threadIdx.x >> 5;
  const int row  = blockIdx.x * WPB + wave;

  float* xr = lds + wave * (2 * PLANE);
  float* xi = xr + PLANE;

  const MFrag A = make_m16(lane);

  const float* dinr = ph;            const float* dini = ph + DIM;
  const float* dthr = ph + 2 * DIM;  const float* dthi = ph + 3 * DIM;
  const float* dphr = ph + 4 * DIM;  const float* dphi = ph + 5 * DIM;

  // (x * d_in) @ bs : three axis passes of M16
  pass1_global(A, x_real + (size_t)row * DIM, x_imag + (size_t)row * DIM,
               dinr, dini, xr, xi, lane);
  lds_pass<256, 16, 1, 16, 1, false>(A, xr, xi, nullptr, nullptr, lane);
  lds_pass<16, 256, 1, 256, 1, false>(A, xr, xi, nullptr, nullptr, lane);
  // (* d_th) @ bs : d_th folded into the first axis pass of the second bs
  lds_pass<256, 1, 16, 1, 16, true>(A, xr, xi, dthr, dthi, lane);
  lds_pass<256, 16, 1, 16, 1, false>(A, xr, xi, nullptr, nullptr, lane);
  // last axis pass + (* d_ph) + interleaved complex64 store
  pass3_global(A, xr, xi, dphr, dphi, out + (size_t)row * 2 * DIM, lane);
}

// Precompute the three diagonal phase tables (cos/sin), 6*4096 floats in ws.
__global__ void phase_setup(const float* __restrict__ alphas,
                            const float* __restrict__ betas,
                            const float* __restrict__ thetas,
                            const float* __restrict__ phis,
                            float* __restrict__ ph) {
  const int c = blockIdx.x * blockDim.x + threadIdx.x;
  if (c >= DIM) return;
  float pin = 0.f, pth = 0.f, pph = 0.f;
#pragma unroll
  for (int k = 0; k < NQ; ++k) {
    const int bit = (c >> (NQ - 1 - k)) & 1;   // qubit 0 = MSB of column index
    pin += bit ? betas[k] : alphas[k];
    pth += bit ? 0.0f : thetas[k];
    pph += bit ? 0.0f : phis[k];
  }
  float s, co;
  sincosf(pin, &s, &co); ph[c]           = co; ph[DIM + c]     = s;
  sincosf(pth, &s, &co); ph[2 * DIM + c] = co; ph[3 * DIM + c] = s;
  sincosf(pph, &s, &co); ph[4 * DIM + c] = co; ph[5 * DIM + c] = s;
}

extern "C" void kernel_launch(void* const* d_in, const int* in_sizes, int n_in,
                              void* d_out, int out_size, void* d_ws,
                              size_t ws_size, hipStream_t stream) {
  const float* x_real = (const float*)d_in[0];
  const float* x_imag = (const float*)d_in[1];
  const float* alphas = (const float*)d_in[2];
  const float* betas  = (const float*)d_in[3];
  const float* thetas = (const float*)d_in[4];
  const float* phis   = (const float*)d_in[5];
  float* ph = (float*)d_ws;  // needs 6*4096*4 = 96 KB of scratch

  phase_setup<<<DIM / 256, 256, 0, stream>>>(alphas, betas, thetas, phis, ph);

  const size_t lds_bytes = (size_t)WPB * 2 * PLANE * sizeof(float);  // ~273 KB
  qphot_kernel<<<BATCH / WPB, 32 * WPB, lds_bytes, stream>>>(
      x_real, x_imag, ph, (float*)d_out);
}